// Odo_position_Transformer_13898514170481
// MI455X (gfx1250) — compile-verified
//
#include <hip/hip_runtime.h>

typedef __attribute__((ext_vector_type(2))) float v2f;
typedef __attribute__((ext_vector_type(8))) float v8f;

#define BATCH 16
#define HDIM  1024
#define WDIM  1024
#define HITCAP 2048

// --- zero the hit counter (in-stream, graph-capture safe) ---
__global__ void odo_init_ws(unsigned int* cnt) {
    if (threadIdx.x == 0 && blockIdx.x == 0) *cnt = 0u;
}

// --- main kernel: one wave computes an 8(h) x 16(w) pixel tile via one
//     V_WMMA_F32_16X16X4_F32.  D = A(16x4) * B(4x16) + C(16x16):
//       A row m, K=0  : m even -> t00, m odd -> t10   (K=1..3 zero)
//       B row K=0, n  : xs(w0+n)                      (K=1..3 zero)
//       C[2i][n]      : fma(t01, ys(h0+i), t02)
//       C[2i+1][n]    : fma(t11, ys(h0+i), t12)
//     => D[2i][n] = gx, D[2i+1][n] = gy for pixel (h0+i, w0+n).
__global__ __launch_bounds__(256) void odo_grid_kernel(
    const float* __restrict__ rot,      // B x 2 x 3
    float* __restrict__ out,            // B x 1 x H x W
    unsigned int* __restrict__ cnt,
    int* __restrict__ hits)
{
    const int lane = threadIdx.x & 31;
    const int wid  = (blockIdx.x * blockDim.x + threadIdx.x) >> 5;

    // tiles per batch: (1024/8) * (1024/16) = 128 * 64 = 8192
    const int b    = wid >> 13;
    const int rem  = wid & 8191;
    const int h0   = (rem >> 6) << 3;   // tile row * 8
    const int w0   = (rem & 63) << 4;   // tile col * 16

    const float t00 = rot[b*6+0], t01 = rot[b*6+1], t02 = rot[b*6+2];
    const float t10 = rot[b*6+3], t11 = rot[b*6+4], t12 = rot[b*6+5];

    const bool hi = (lane >= 16);
    const int  n  = lane & 15;

    // A operand: 16x4 f32, only K=0 (VGPR0, lanes 0-15) non-zero.
    v2f A;
    A.x = hi ? 0.0f : ((lane & 1) ? t10 : t00);
    A.y = 0.0f;

    // B operand: 4x16 f32, only K=0 row (VGPR0, lanes 0-15) non-zero.
    const float xs = ((2.0f * (float)(w0 + n) + 1.0f) / 1024.0f) - 1.0f;
    v2f Bm;
    Bm.x = hi ? 0.0f : xs;
    Bm.y = 0.0f;

    // C operand: 16x16 f32 accumulator layout (doc-verified):
    //   lanes 0-15 : VGPR r -> M=r ; lanes 16-31 : VGPR r -> M=8+r
    v8f C;
#pragma unroll
    for (int r = 0; r < 8; ++r) {
        const int   i  = (r >> 1) + (hi ? 4 : 0);
        const float ys = ((2.0f * (float)(h0 + i) + 1.0f) / 1024.0f) - 1.0f;
        C[r] = (r & 1) ? __builtin_fmaf(t11, ys, t12)
                       : __builtin_fmaf(t01, ys, t02);
    }

    // D = A*B + C  (EXEC is all-ones: exact grid, no divergence before here)
    v8f D = __builtin_amdgcn_wmma_f32_16x16x4_f32(
        /*neg_a=*/false, A, /*neg_b=*/false, Bm,
        /*c_mod=*/(short)0, C, /*reuse_a=*/false, /*reuse_b=*/false);

    const int w = w0 + n;
#pragma unroll
    for (int k = 0; k < 4; ++k) {
        const float gx = D[2*k];
        const float gy = D[2*k + 1];
        const int   h  = h0 + k + (hi ? 4 : 0);

        // nearest + border clip; 200/500 are interior so clip can't create a hit
        const float ixf = ((gx + 1.0f) * 1024.0f - 1.0f) * 0.5f;
        const float iyf = ((gy + 1.0f) * 1024.0f - 1.0f) * 0.5f;
        const bool hit = (__builtin_rintf(ixf) == 200.0f) &&
                         (__builtin_rintf(iyf) == 500.0f);

        const int flat = (((b << 10) + h) << 10) + w;  // b*H*W + h*W + w
        out[flat] = hit ? 30.0f : 0.0f;

        if (hit) {
            unsigned int slot = atomicAdd(cnt, 1u);
            if (slot < HITCAP) hits[slot] = flat;
        }
    }
}

// --- finalize: reproduce argwhere ordering (ascending flat index),
//     first 64 hits -> (h, w) as floats, pad with -1 ---
__global__ void odo_finalize_kernel(const unsigned int* __restrict__ cnt,
                                    const int* __restrict__ hits,
                                    float* __restrict__ odo)
{
    __shared__ int n_s;
    const int tid = threadIdx.x;
    if (tid < 128) odo[tid] = -1.0f;
    if (tid == 0) {
        unsigned int c = *cnt;
        n_s = (c > (unsigned int)HITCAP) ? HITCAP : (int)c;
    }
    __syncthreads();
    const int nn = n_s;
    for (int j = tid; j < nn; j += blockDim.x) {
        const int my = hits[j];
        int rank = 0;
        for (int k = 0; k < nn; ++k) rank += (hits[k] < my) ? 1 : 0;
        if (rank < 64) {
            odo[rank*2]     = (float)((my >> 10) & 1023);  // h
            odo[rank*2 + 1] = (float)(my & 1023);          // w
        }
    }
}

extern "C" void kernel_launch(void* const* d_in, const int* in_sizes, int n_in,
                              void* d_out, int out_size, void* d_ws, size_t ws_size,
                              hipStream_t stream)
{
    // d_in[0] = cost_map_batch (shape-only, unused), d_in[1] = rot_mat (B x 2 x 3)
    const float* rot = (const float*)d_in[1];
    float* out = (float*)d_out;
    float* odo = out + (size_t)BATCH * HDIM * WDIM;   // 128 floats after the grid

    unsigned int* cnt  = (unsigned int*)d_ws;
    int*          hits = (int*)((char*)d_ws + 16);

    odo_init_ws<<<1, 1, 0, stream>>>(cnt);

    const int waves  = BATCH * (HDIM / 8) * (WDIM / 16); // 131072 waves
    const int blocks = waves / 8;                        // 256 thr = 8 waves
    odo_grid_kernel<<<blocks, 256, 0, stream>>>(rot, out, cnt, hits);

    odo_finalize_kernel<<<1, 128, 0, stream>>>(cnt, hits, odo);
}